// MappedAvgUnpool_34282428956674
// MI455X (gfx1250) — compile-verified
//
#include <hip/hip_runtime.h>
#include <stdint.h>

// Problem constants (fixed by the reference setup)
#define KSZ   4          // kernel taps
#define NIP   4          // interp points per tap
#define TAPS  (KSZ*NIP)  // 16 scatter points per input pixel
#define PIX   256        // input pixels staged per block
#define NBC   16         // (b,c) slices processed per block (reuse of LDS stage)

// Scatter-add unpool:
//   out[bc, y*ow+x] += x[bc, n] * w[n,t]/K   for each of 16 taps t of pixel n
// Block = 256 threads: stages 256 pixels' maps+weights into LDS (transposed,
// tap-major) via CDNA5 async global->LDS DMA, then loops 16 channel slices,
// each thread owning one pixel and firing 16 global_atomic_add_f32 per slice.
__global__ __launch_bounds__(256) void
MappedAvgUnpool_scatter(const float* __restrict__ x,
                        const int*   __restrict__ smap,   // (IH,IW,K,P,2) -> (x,y)
                        const float* __restrict__ w,      // (IH,IW,K,P)
                        const int*   __restrict__ ohp,
                        const int*   __restrict__ owp,
                        float*       __restrict__ out,
                        int N)                             // IH*IW
{
    __shared__ int2  map_lds[TAPS][PIX];   // 32 KB, tap-major (conflict-free reads)
    __shared__ float w_lds[TAPS][PIX];     // 16 KB

    const int tid = threadIdx.x;           // 0..255 : local pixel id in compute phase
    const int n0  = blockIdx.x * PIX;      // first input pixel of this block
    const int bc0 = blockIdx.y * NBC;      // first (b,c) slice of this block

    const int ow = owp[0];
    const int os = ohp[0] * ow;            // output pixels per (b,c) slice

    // ---- Stage maps + weights into LDS with async DMA (ASYNCcnt path) ----
    // Global layout is pixel-major [p][t]; we scatter into LDS tap-major
    // [t][p] using the per-lane LDS address of the async instructions.
    const uint32_t lm = (uint32_t)(size_t)(&map_lds[0][0]);  // LDS byte offset
    const uint32_t lw = (uint32_t)(size_t)(&w_lds[0][0]);
    const uint64_t gm = (uint64_t)(size_t)smap + (uint64_t)n0 * (TAPS * 8u);
    const uint64_t gw = (uint64_t)(size_t)w    + (uint64_t)n0 * (TAPS * 4u);
#pragma unroll
    for (int it = 0; it < TAPS; ++it) {
        const int e = it * PIX + tid;          // entry id: lanes are consecutive
        const int p = e >> 4;                  // local pixel
        const int t = e & (TAPS - 1);          // tap
        const uint32_t dm = lm + (uint32_t)(t * PIX + p) * 8u;
        const uint64_t sm = gm + (uint64_t)e * 8u;   // coalesced 256B/wave
        asm volatile("global_load_async_to_lds_b64 %0, %1, off"
                     :: "v"(dm), "v"(sm) : "memory");
        const uint32_t dw = lw + (uint32_t)(t * PIX + p) * 4u;
        const uint64_t sw = gw + (uint64_t)e * 4u;
        asm volatile("global_load_async_to_lds_b32 %0, %1, off"
                     :: "v"(dw), "v"(sw) : "memory");
    }
    asm volatile("s_wait_asynccnt 0" ::: "memory");
    __syncthreads();

    // ---- Scatter phase: reuse the staged tile across NBC channel slices ----
    // Unroll x2: two independent x-loads + two 16-atomic bursts in flight.
#pragma unroll 1
    for (int s = 0; s < NBC; s += 2) {
        const int bc  = bc0 + s;
        const float xv0 = 0.25f * x[(size_t)bc       * (size_t)N + (size_t)(n0 + tid)];
        const float xv1 = 0.25f * x[(size_t)(bc + 1) * (size_t)N + (size_t)(n0 + tid)];
        float* outb0 = out + (size_t)bc       * (size_t)os;
        float* outb1 = out + (size_t)(bc + 1) * (size_t)os;
#pragma unroll
        for (int t = 0; t < TAPS; ++t) {
            const int2  m  = map_lds[t][tid];    // ds_load_2addr pairs, conflict-free
            const float wv = w_lds[t][tid];
            const int   idx = m.y * ow + m.x;
            (void)__hip_atomic_fetch_add(outb0 + idx, xv0 * wv,
                                         __ATOMIC_RELAXED,
                                         __HIP_MEMORY_SCOPE_AGENT);
            (void)__hip_atomic_fetch_add(outb1 + idx, xv1 * wv,
                                         __ATOMIC_RELAXED,
                                         __HIP_MEMORY_SCOPE_AGENT);
        }
    }
}

extern "C" void kernel_launch(void* const* d_in, const int* in_sizes, int n_in,
                              void* d_out, int out_size, void* d_ws, size_t ws_size,
                              hipStream_t stream)
{
    const float* x    = (const float*)d_in[0];
    const int*   smap = (const int*)  d_in[1];
    const float* w    = (const float*)d_in[2];
    const int*   ohp  = (const int*)  d_in[3];
    const int*   owp  = (const int*)  d_in[4];
    float*       out  = (float*)d_out;

    const int N  = in_sizes[1] / (KSZ * NIP * 2);  // IH*IW = 32768
    const int BC = in_sizes[0] / N;                // B*C   = 512

    // Zero-init the 256 MB output (scatter targets are sparse/random).
    hipMemsetAsync(d_out, 0, (size_t)out_size * sizeof(float), stream);

    dim3 grid((unsigned)(N / PIX), (unsigned)(BC / NBC));  // 128 x 32 = 4096 blocks
    MappedAvgUnpool_scatter<<<grid, PIX, 0, stream>>>(x, smap, w, ohp, owp, out, N);
}